// MaskedNorm_52931176956564
// MI455X (gfx1250) — compile-verified
//
#include <hip/hip_runtime.h>

typedef float v2f __attribute__((ext_vector_type(2)));
typedef float v4f __attribute__((ext_vector_type(4)));
typedef float v8f __attribute__((ext_vector_type(8)));

#define FEAT   256
#define EPSV   1e-5f
// ws float layout
#define WS_SUM   0
#define WS_SQ    256
#define WS_CNT   512
#define WS_MEAN  516   // 16B aligned (516*4 = 2064)
#define WS_ISTD  772   // 16B aligned (772*4 = 3088)

// ---------------------------------------------------------------------------
// Kernel 1: masked partial sums of x and x^2 per feature.
// Layout: 256 threads = 64 feature-groups (float4) x 4 row-lanes.
// All loads are b128, fully coalesced; x is read with default (RT) temporal
// hints so it stays resident in the 192MB L2 for the apply pass.
// Block-tail 4-way reduction is done with V_WMMA_F32_16X16X4_F32 (ones x B).
// ---------------------------------------------------------------------------
__global__ __launch_bounds__(256) void mn_partial_k(const float* __restrict__ x,
                                                    const int*   __restrict__ mask,
                                                    float*       __restrict__ ws,
                                                    int nrows) {
    __shared__ float s_sum[4][FEAT];
    __shared__ float s_sq [4][FEAT];

    const int tid = threadIdx.x;
    const int fg  = tid & 63;     // feature group (4 features)
    const int r   = tid >> 6;     // row lane 0..3
    const int f0  = fg * 4;

    v4f acc  = {0.f, 0.f, 0.f, 0.f};
    v4f accq = {0.f, 0.f, 0.f, 0.f};
    float cnt = 0.f;

    const int stride = gridDim.x * 4;
    for (int row = blockIdx.x * 4 + r; row < nrows; row += stride) {
        const float* xp = x + (size_t)row * FEAT + f0;
        __builtin_prefetch(xp + (size_t)stride * FEAT, 0, 1);  // global_prefetch_b8 -> L2
        v4f v = *(const v4f*)xp;                               // global_load_b128
        float m = (mask[row] > 0) ? 1.f : 0.f;
        v4f t = v * m;
        acc  += t;
        accq += v * t;
        if (fg == 0) cnt += m;  // count each row exactly once
    }

    s_sum[r][f0+0] = acc.x;  s_sum[r][f0+1] = acc.y;
    s_sum[r][f0+2] = acc.z;  s_sum[r][f0+3] = acc.w;
    s_sq [r][f0+0] = accq.x; s_sq [r][f0+1] = accq.y;
    s_sq [r][f0+2] = accq.z; s_sq [r][f0+3] = accq.w;
    __syncthreads();

    // --- WMMA tail reduction: D = ones(16x4) x B(4x16) sums the 4 row-lanes.
    // All 16 rows of A are identical, so every lane's d[0] is the column sum
    // for feature (fb + hi*16 + ln16) irrespective of the exact K striping.
    const int wv   = tid >> 5;          // wave 0..7 -> features wv*32..wv*32+31
    const int ln16 = tid & 15;
    const int hi   = (tid >> 4) & 1;    // lane half
    const int k0   = hi ? 2 : 0;        // K rows supplied by this lane half
    const int fb   = wv * 32;

    v2f a; a.x = 1.f; a.y = 1.f;        // ones A-matrix (16x4 f32 = 2 VGPRs)
    v8f c = {};

    v2f bs0, bq0, bs1, bq1;
    bs0.x = s_sum[k0  ][fb + ln16];       bs0.y = s_sum[k0+1][fb + ln16];
    bq0.x = s_sq [k0  ][fb + ln16];       bq0.y = s_sq [k0+1][fb + ln16];
    bs1.x = s_sum[k0  ][fb + 16 + ln16];  bs1.y = s_sum[k0+1][fb + 16 + ln16];
    bq1.x = s_sq [k0  ][fb + 16 + ln16];  bq1.y = s_sq [k0+1][fb + 16 + ln16];

    v8f ds0 = __builtin_amdgcn_wmma_f32_16x16x4_f32(false, a, false, bs0, (short)0, c, false, false);
    v8f dq0 = __builtin_amdgcn_wmma_f32_16x16x4_f32(false, a, false, bq0, (short)0, c, false, false);
    v8f ds1 = __builtin_amdgcn_wmma_f32_16x16x4_f32(false, a, false, bs1, (short)0, c, false, false);
    v8f dq1 = __builtin_amdgcn_wmma_f32_16x16x4_f32(false, a, false, bq1, (short)0, c, false, false);

    const int f = fb + (hi ? 16 : 0) + ln16;
    float s = hi ? ds1[0] : ds0[0];
    float q = hi ? dq1[0] : dq0[0];
    atomicAdd(&ws[WS_SUM + f], s);       // global_atomic_add_f32
    atomicAdd(&ws[WS_SQ  + f], q);

    if (fg == 0) atomicAdd(&ws[WS_CNT], cnt);
}

// ---------------------------------------------------------------------------
// Kernel 2: finalize mean / invstd per feature (tiny).
// ---------------------------------------------------------------------------
__global__ __launch_bounds__(256) void mn_finalize_k(float* __restrict__ ws) {
    const int f = threadIdx.x;
    const float cnt  = ws[WS_CNT];
    const float mean = ws[WS_SUM + f] / cnt;
    float var = ws[WS_SQ + f] / cnt - mean * mean;
    var = fmaxf(var, 0.f);
    ws[WS_MEAN + f] = mean;
    ws[WS_ISTD + f] = rsqrtf(var + EPSV);
}

// ---------------------------------------------------------------------------
// Kernel 3: apply. Reads x (expected L2-resident after kernel 1), writes out
// with non-temporal b128 stores so the output stream never evicts x from L2.
// y = x*scale + shift for masked rows, passthrough otherwise.
// ---------------------------------------------------------------------------
__global__ __launch_bounds__(256) void mn_apply_k(const float* __restrict__ x,
                                                  const int*   __restrict__ mask,
                                                  const float* __restrict__ gamma,
                                                  const float* __restrict__ beta,
                                                  const float* __restrict__ ws,
                                                  float*       __restrict__ out,
                                                  int nrows) {
    const int tid = threadIdx.x;
    const int fg  = tid & 63;
    const int r   = tid >> 6;
    const int f0  = fg * 4;

    v4f g  = *(const v4f*)(gamma + f0);
    v4f bb = *(const v4f*)(beta  + f0);
    v4f mu = *(const v4f*)(ws + WS_MEAN + f0);
    v4f is = *(const v4f*)(ws + WS_ISTD + f0);
    v4f sc = is * g;            // scale = invstd*gamma
    v4f sh = bb - mu * sc;      // shift = beta - mean*scale

    const int stride = gridDim.x * 4;
    for (int row = blockIdx.x * 4 + r; row < nrows; row += stride) {
        const size_t off = (size_t)row * FEAT + f0;
        __builtin_prefetch(x + off + (size_t)stride * FEAT, 0, 1);
        v4f v = *(const v4f*)(x + off);              // global_load_b128 (L2 hit)
        v4f y = v * sc + sh;
        if (!(mask[row] > 0)) y = v;                  // masked_scatter semantics
        __builtin_nontemporal_store(y, (v4f*)(out + off));  // NT b128 store
    }
}

// ---------------------------------------------------------------------------
extern "C" void kernel_launch(void* const* d_in, const int* in_sizes, int n_in,
                              void* d_out, int out_size, void* d_ws, size_t ws_size,
                              hipStream_t stream) {
    const float* x     = (const float*)d_in[0];
    const int*   mask  = (const int*)  d_in[1];
    const float* gamma = (const float*)d_in[2];
    const float* beta  = (const float*)d_in[3];
    float*       out   = (float*)d_out;
    float*       ws    = (float*)d_ws;

    const int nrows = in_sizes[1];   // B*T = 131072

    // zero the accumulation region (sum, sumsq, count) every call (graph-safe)
    hipMemsetAsync(d_ws, 0, (WS_CNT + 1) * sizeof(float), stream);

    mn_partial_k <<<256,  256, 0, stream>>>(x, mask, ws, nrows);
    mn_finalize_k<<<1,    256, 0, stream>>>(ws);
    mn_apply_k   <<<1024, 256, 0, stream>>>(x, mask, gamma, beta, ws, out, nrows);
}